// VariationalGCNEncoder_9491877724563
// MI455X (gfx1250) — compile-verified
//
#include <hip/hip_runtime.h>
#include <hip/hip_bf16.h>

#define NNODES 50000
#define NEDGES 800000
#define KDIM   256
#define KBLK   64            // K-block staged in LDS
#define NW     8             // waves per block
#define PS     160           // LDS pair-row stride (floats): PS%64==32 -> conflict-free halves

typedef __attribute__((ext_vector_type(2))) float v2f;
typedef __attribute__((ext_vector_type(8))) float v8f;

// ---------------- degree / inverse-degree ----------------
__global__ void deg_kernel(const long long* __restrict__ ei, float* __restrict__ deg, int E) {
    int gid = blockIdx.x * blockDim.x + threadIdx.x;
    int stride = gridDim.x * blockDim.x;
    for (int e = gid; e < E; e += stride) {
        int d = (int)ei[(long long)E + e];
        atomicAdd(&deg[d], 1.0f);
    }
}

__global__ void invdeg_kernel(float* __restrict__ deg, int n) {
    int gid = blockIdx.x * blockDim.x + threadIdx.x;
    if (gid < n) deg[gid] = 1.0f / fmaxf(deg[gid], 1.0f);
}

// ---------------- edge scatter-add (mean numerator) ----------------
__global__ __launch_bounds__(256) void scatter_add_kernel(
        const float* __restrict__ x, const long long* __restrict__ ei,
        float* __restrict__ agg, int E) {
    int gid    = blockIdx.x * blockDim.x + threadIdx.x;
    int lane   = gid & 31;
    int wave   = gid >> 5;
    int nwaves = (gridDim.x * blockDim.x) >> 5;
    for (int e = wave; e < E; e += nwaves) {
        long long s = ei[e];
        long long d = ei[E + e];
        int e2 = e + nwaves;
        if (e2 < E) {
            long long s2 = ei[e2];
            __builtin_prefetch(x + s2 * (long long)KDIM + lane * 8, 0, 0);
        }
        const float* xs = x + s * (long long)KDIM;
        float*       ad = agg + d * (long long)KDIM;
#pragma unroll
        for (int f0 = 0; f0 < KDIM; f0 += 32) {
            atomicAdd(&ad[f0 + lane], xs[f0 + lane]);
        }
    }
}

// ---------------- fused SAGE linear: out = (agg*inv_deg)@Wl + x@Wr + b ----------------
// fp32 WMMA 16x16x4. Each wave computes a 16x64 output slab (4 accumulators).
// Weight panels stored K-pair-interleaved in LDS so each B fragment is one
// aligned ds_load_b64 directly into an even VGPR pair (no repack movs).
__global__ __launch_bounds__(256) void sage_gemm_kernel(
        const float* __restrict__ agg, const float* __restrict__ xin,
        const float* __restrict__ invdeg,
        const float* __restrict__ Wl, const float* __restrict__ Wr,
        const float* __restrict__ bias, float* __restrict__ out,
        int nrows, int ncols, int do_relu) {
    __shared__ float sWl[(KBLK / 2) * PS];   // 20 KB
    __shared__ float sWr[(KBLK / 2) * PS];   // 20 KB

    const int tid  = threadIdx.x;
    const int col0 = blockIdx.y * 64;           // 64-wide column slab per block
    const int wave = tid >> 5;
    const int lane = tid & 31;
    const int half = lane >> 4;                 // 0: K pair {0,1}, 1: K pair {2,3}
    const int m    = lane & 15;

    const int  t      = blockIdx.x * NW + wave; // 16-row tile index
    const bool active = t < (nrows >> 4);
    const int  row0   = active ? t * 16 : 0;    // clamp so barriers stay uniform
    const int  row    = row0 + m;

    const float idg = invdeg[row];
    const float2* aggp = (const float2*)(agg + (size_t)row * KDIM);
    const float2* xp   = (const float2*)(xin + (size_t)row * KDIM);

    v8f acc0 = {}, acc1 = {}, acc2 = {}, acc3 = {};

    for (int kb0 = 0; kb0 < KDIM; kb0 += KBLK) {
        __syncthreads();
        // stage KBLK x 64 weight panels, K-pair interleaved:
        // sW[p*PS + 2c + j] = W[kb0 + 2p + j][col0 + c]
        for (int i = tid; i < (KBLK / 2) * 64; i += 256) {
            int p = i >> 6, c = i & 63;
            size_t g = (size_t)(kb0 + 2 * p) * ncols + col0 + c;
            float2 wl, wr;
            wl.x = Wl[g];         wl.y = Wl[g + ncols];
            wr.x = Wr[g];         wr.y = Wr[g + ncols];
            *(float2*)&sWl[p * PS + 2 * c] = wl;
            *(float2*)&sWr[p * PS + 2 * c] = wr;
        }
        __syncthreads();

#pragma unroll 4
        for (int k = 0; k < KBLK; k += 4) {
            int ka = ((kb0 + k) >> 1) + half;   // float2 index of A pair
            float2 a2 = aggp[ka];
            float2 x2 = xp[ka];
            v2f afrag; afrag[0] = a2.x * idg; afrag[1] = a2.y * idg;
            v2f xfrag; xfrag[0] = x2.x;       xfrag[1] = x2.y;

            int pb = ((k >> 1) + half) * PS + 2 * m;   // pair-row base in LDS
#define TILE_MAC(ACC, C0)                                                           \
            {                                                                       \
                v2f bl = *(const v2f*)&sWl[pb + 2 * (C0)];                          \
                v2f br = *(const v2f*)&sWr[pb + 2 * (C0)];                          \
                ACC = __builtin_amdgcn_wmma_f32_16x16x4_f32(false, afrag, false, bl,\
                                                            (short)0, ACC, false, false);\
                ACC = __builtin_amdgcn_wmma_f32_16x16x4_f32(false, xfrag, false, br,\
                                                            (short)0, ACC, false, false);\
            }
            TILE_MAC(acc0, 0)
            TILE_MAC(acc1, 16)
            TILE_MAC(acc2, 32)
            TILE_MAC(acc3, 48)
#undef TILE_MAC
        }
    }

    if (active) {
        v8f* accs[4] = {&acc0, &acc1, &acc2, &acc3};
#pragma unroll
        for (int tt = 0; tt < 4; ++tt) {
            const float bv = bias[col0 + tt * 16 + m];
            v8f a = *accs[tt];
#pragma unroll
            for (int i = 0; i < 8; ++i) {
                int r = row0 + half * 8 + i;
                float v = a[i] + bv;
                if (do_relu) v = (v > 0.0f) ? v : 0.01f * v;   // leaky_relu 0.01
                out[(size_t)r * ncols + col0 + tt * 16 + m] = v;
            }
        }
    }
}

// ---------------- host orchestration ----------------
extern "C" void kernel_launch(void* const* d_in, const int* in_sizes, int n_in,
                              void* d_out, int out_size, void* d_ws, size_t ws_size,
                              hipStream_t stream) {
    (void)in_sizes; (void)n_in; (void)out_size; (void)ws_size;

    const float*     x  = (const float*)d_in[0];
    const long long* ei = (const long long*)d_in[1];   // int64 edge_index [2,E]
    const float* W[6][3];                              // l0,l1,l2,c2,mu,ls : Wl,Wr,b
    for (int l = 0; l < 6; ++l)
        for (int j = 0; j < 3; ++j)
            W[l][j] = (const float*)d_in[2 + l * 3 + j];

    float* wsf     = (float*)d_ws;
    float* invdeg  = wsf;                                   // N floats
    float* bufA    = wsf + 50176;                           // N x 256
    float* bufAgg  = bufA  + (size_t)NNODES * KDIM;         // N x 256
    float* bufB    = bufAgg + (size_t)NNODES * KDIM;        // N x 256
    float* outmu   = (float*)d_out;                         // N x 128
    float* outls   = outmu + (size_t)NNODES * 128;          // N x 128

    hipMemsetAsync(invdeg, 0, NNODES * sizeof(float), stream);
    deg_kernel<<<512, 256, 0, stream>>>(ei, invdeg, NEDGES);
    invdeg_kernel<<<(NNODES + 255) / 256, 256, 0, stream>>>(invdeg, NNODES);

    const int ntiles = NNODES / 16;                       // 3125 (exact)
    dim3 g256((ntiles + NW - 1) / NW, 256 / 64);
    dim3 g128((ntiles + NW - 1) / NW, 128 / 64);

    auto layer = [&](const float* in, const float* Wl, const float* Wr, const float* b,
                     float* outbuf, int ncols, int relu) {
        hipMemsetAsync(bufAgg, 0, (size_t)NNODES * KDIM * sizeof(float), stream);
        scatter_add_kernel<<<1024, 256, 0, stream>>>(in, ei, bufAgg, NEDGES);
        sage_gemm_kernel<<<(ncols == 256 ? g256 : g128), 256, 0, stream>>>(
            bufAgg, in, invdeg, Wl, Wr, b, outbuf, NNODES, ncols, relu);
    };

    layer(x,    W[0][0], W[0][1], W[0][2], bufB, 256, 1);   // l0 + leaky_relu
    layer(bufB, W[1][0], W[1][1], W[1][2], bufA, 256, 1);   // l1 + leaky_relu
    layer(bufA, W[2][0], W[2][1], W[2][2], bufB, 256, 1);   // l2 + leaky_relu
    layer(bufB, W[3][0], W[3][1], W[3][2], bufA, 256, 0);   // c2 (no relu)

    // mu and logstd share one aggregation of h = bufA
    hipMemsetAsync(bufAgg, 0, (size_t)NNODES * KDIM * sizeof(float), stream);
    scatter_add_kernel<<<1024, 256, 0, stream>>>(bufA, ei, bufAgg, NEDGES);
    sage_gemm_kernel<<<g128, 256, 0, stream>>>(bufAgg, bufA, invdeg,
                                               W[4][0], W[4][1], W[4][2], outmu, NNODES, 128, 0);
    sage_gemm_kernel<<<g128, 256, 0, stream>>>(bufAgg, bufA, invdeg,
                                               W[5][0], W[5][1], W[5][2], outls, NNODES, 128, 0);
}